// PiecewiseActivation_36704790512130
// MI455X (gfx1250) — compile-verified
//
#include <hip/hip_runtime.h>
#include <hip/hip_bf16.h>
#include <stdint.h>

#define N_KNOTS 16
#define TPB 256
#define TILE_BYTES (TPB * 16)   // one float4 per thread per tile = 4 KB

// ---- CDNA5 async global->LDS copy (gfx1250). Each active lane copies 16B
// from its global address to its LDS byte offset. Tracked by ASYNCcnt.
__device__ __forceinline__ void async_copy_b128(const float4* gsrc, uint32_t lds_off) {
    asm volatile("global_load_async_to_lds_b128 %0, %1, off"
                 :: "v"(lds_off), "v"(gsrc)
                 : "memory");
}

// Piecewise-linear eval using the packed LDS knot table:
//   tab[i] = { xs[i], ys[i], (ys[i+1]-ys[i])/(xs[i+1]-xs[i]), xs[i+1] }
__device__ __forceinline__ float pwl_eval(float x, const float4* tab,
                                          float xs0, float xs15,
                                          float ys0, float ys15,
                                          float sL, float sR, float inv_h) {
    float t = (x - xs0) * inv_h;
    int i = (int)floorf(t);
    i = i < 0 ? 0 : i;
    i = i > (N_KNOTS - 2) ? (N_KNOTS - 2) : i;
    float4 kn = tab[i];                       // ds_load_b128
    if (x < kn.x && i > 0)                  { --i; kn = tab[i]; }
    else if (x >= kn.w && i < N_KNOTS - 2)  { ++i; kn = tab[i]; }
    float r = kn.y + (x - kn.x) * kn.z;
    if (x < xs0)  r = ys0 - (xs0 - x) * sL;   // left extrapolation
    if (x > xs15) r = ys15 + (x - xs15) * sR; // right extrapolation
    return r;
}

__global__ __launch_bounds__(TPB) void PiecewiseActivation_kernel(
    const float* __restrict__ x, const float* __restrict__ xs,
    const float* __restrict__ slopes, const float* __restrict__ ys,
    float* __restrict__ out, int n, int nv4, int tiles) {

    __shared__ float4 s_buf[2][TPB];   // double-buffered async staging (8 KB)
    __shared__ float4 s_tab[N_KNOTS];  // packed knot table

    const int tid = threadIdx.x;

    // Build knot table once per block (15 divides per block total).
    if (tid < N_KNOTS) {
        int ip = (tid < N_KNOTS - 1) ? tid + 1 : tid;
        float xi = xs[tid], yi = ys[tid];
        float xn = xs[ip],  yn = ys[ip];
        float d  = (tid < N_KNOTS - 1) ? (yn - yi) / (xn - xi) : 0.0f;
        s_tab[tid] = make_float4(xi, yi, d, xn);
    }
    __syncthreads();

    const float xs0  = xs[0],     xs15 = xs[N_KNOTS - 1];
    const float ys0  = ys[0],     ys15 = ys[N_KNOTS - 1];
    const float sL   = slopes[0], sR   = slopes[1];
    const float inv_h = (float)(N_KNOTS - 1) / (xs15 - xs0);

    const float4* xv4 = (const float4*)x;
    float4*       ov4 = (float4*)out;

    // LDS byte offset of this lane's staging slot (flat-addr low 32 bits == LDS offset).
    const uint32_t lds0 = (uint32_t)(uintptr_t)(&s_buf[0][0]) + (uint32_t)tid * 16u;

    const int t0 = blockIdx.x;
    const int stride = gridDim.x;

    // Prologue: stage first tile.
    if (t0 < tiles) {
        long v4 = (long)t0 * TPB + tid;
        if (v4 >= nv4) v4 = nv4 - 1;            // clamp: always issue (exact ASYNCcnt bookkeeping)
        async_copy_b128(xv4 + v4, lds0);
    }

    int k = 0;
    for (int t = t0; t < tiles; t += stride, ++k) {
        const int tn = t + stride;
        const bool more = tn < tiles;
        if (more) {                              // stage next tile into other buffer
            long v4n = (long)tn * TPB + tid;
            if (v4n >= nv4) v4n = nv4 - 1;
            async_copy_b128(xv4 + v4n, lds0 + (uint32_t)((k + 1) & 1) * (uint32_t)TILE_BYTES);
        }
        if (more) asm volatile("s_wait_asynccnt 0x1" ::: "memory"); // current tile landed
        else      asm volatile("s_wait_asynccnt 0x0" ::: "memory"); // drain

        float4 v = s_buf[k & 1][tid];            // ds_load_b128 from staged tile
        float4 r;
        r.x = pwl_eval(v.x, s_tab, xs0, xs15, ys0, ys15, sL, sR, inv_h);
        r.y = pwl_eval(v.y, s_tab, xs0, xs15, ys0, ys15, sL, sR, inv_h);
        r.z = pwl_eval(v.z, s_tab, xs0, xs15, ys0, ys15, sL, sR, inv_h);
        r.w = pwl_eval(v.w, s_tab, xs0, xs15, ys0, ys15, sL, sR, inv_h);

        long v4 = (long)t * TPB + tid;
        if (v4 < nv4) ov4[v4] = r;               // global_store_b128
    }

    // Tail elements (n % 4), direct path, block 0 only.
    const int rem = n - nv4 * 4;
    if (blockIdx.x == 0 && tid < rem) {
        int idx = nv4 * 4 + tid;
        out[idx] = pwl_eval(x[idx], s_tab, xs0, xs15, ys0, ys15, sL, sR, inv_h);
    }
}

extern "C" void kernel_launch(void* const* d_in, const int* in_sizes, int n_in,
                              void* d_out, int out_size, void* d_ws, size_t ws_size,
                              hipStream_t stream) {
    const float* x      = (const float*)d_in[0];
    const float* xs     = (const float*)d_in[1];
    const float* slopes = (const float*)d_in[2];
    const float* ys     = (const float*)d_in[3];
    float* out = (float*)d_out;

    const int n     = in_sizes[0];               // 8*1024*1024
    const int nv4   = n >> 2;
    const int tiles = (nv4 + TPB - 1) / TPB;     // 8192 for the reference size
    int blocks = tiles < 1024 ? (tiles < 1 ? 1 : tiles) : 1024;  // ~8 pipelined tiles/block

    PiecewiseActivation_kernel<<<blocks, TPB, 0, stream>>>(x, xs, slopes, ys, out, n, nv4, tiles);
}